// TreeLSTM_17042430230604
// MI455X (gfx1250) — compile-verified
//
#include <hip/hip_runtime.h>

typedef float v2f __attribute__((ext_vector_type(2)));
typedef float v8f __attribute__((ext_vector_type(8)));

constexpr int kS = 512;   // sequence / tree length
constexpr int kB = 128;   // batch
constexpr int kD = 512;   // input dim
constexpr int kH = 256;   // hidden dim
constexpr int kG = 3 * kH;         // 768 gate columns
constexpr int kKC = kD + kH;       // 768 combined K for [x | h] GEMMs
constexpr int kBT = 32;            // batch rows per workgroup (two WMMA M tiles)
constexpr int kThreads = 512;      // 16 waves of 32

// LDS row strides (floats), padded by 4 so 16 A-rows hit 16 distinct banks;
// all strides are multiples of 4 floats -> float4-aligned rows.
constexpr int XS_LD  = kG + 4;   // 772 : X = [input | h_t], later reused for g
constexpr int XFS_LD = kD + 4;   // 516 : gathered parent input rows
constexpr int HS_LD  = kH + 4;   // 260 : h, c, fct buffers

__device__ __forceinline__ float sigmoidf(float x) {
    return 1.0f / (1.0f + __expf(-x));
}

__device__ __forceinline__ v8f wmma_f32(v2f a, v2f b, v8f c) {
    // V_WMMA_F32_16X16X4_F32 : D = A(16x4) * B(4x16) + C(16x16), fp32
    return __builtin_amdgcn_wmma_f32_16x16x4_f32(
        /*neg_a=*/false, a, /*neg_b=*/false, b,
        /*c_mod=*/(short)0, c, /*reuse_a=*/false, /*reuse_b=*/false);
}

extern "C" __global__ void zero_ws_kernel(float4* __restrict__ p, size_t n4) {
    size_t i = (size_t)blockIdx.x * blockDim.x + threadIdx.x;
    size_t stride = (size_t)gridDim.x * blockDim.x;
    float4 z = make_float4(0.f, 0.f, 0.f, 0.f);
    for (; i < n4; i += stride) p[i] = z;
}

// Pack [W_iou;U_iou] (768x768) and [W_f;U_f] (768x256) into WMMA-fragment
// order: packed[kb4][n][q] = W[kb4*4+q][n]. A lane's B fragment (K=kb4*4,
// rows 2*half and 2*half+1 at column n) is then one contiguous 8-byte load.
extern "C" __global__ void pack_weights_kernel(const float* __restrict__ W_iou,
                                               const float* __restrict__ U_iou,
                                               const float* __restrict__ W_f,
                                               const float* __restrict__ U_f,
                                               float* __restrict__ wc,
                                               float* __restrict__ wf) {
    int idx = blockIdx.x * blockDim.x + threadIdx.x;
    const int total_c = (kKC / 4) * kG;   // 192*768
    const int total_f = (kKC / 4) * kH;   // 192*256
    if (idx < total_c) {
        int kb4 = idx / kG, n = idx % kG;
        #pragma unroll
        for (int q = 0; q < 4; ++q) {
            int k = kb4 * 4 + q;
            float v = (k < kD) ? W_iou[(size_t)k * kG + n]
                               : U_iou[(size_t)(k - kD) * kG + n];
            wc[((size_t)kb4 * kG + n) * 4 + q] = v;
        }
    }
    if (idx < total_f) {
        int kb4 = idx / kH, n = idx % kH;
        #pragma unroll
        for (int q = 0; q < 4; ++q) {
            int k = kb4 * 4 + q;
            float v = (k < kD) ? W_f[(size_t)k * kH + n]
                               : U_f[(size_t)(k - kD) * kH + n];
            wf[((size_t)kb4 * kH + n) * 4 + q] = v;
        }
    }
}

extern "C" __global__ __launch_bounds__(kThreads, 1)
void tree_lstm_scan(const float* __restrict__ inputs,   // (S,B,D)
                    const int*   __restrict__ parents,  // (B,S)
                    const float* __restrict__ b_iou,    // (3H)
                    const float* __restrict__ b_f,      // (H)
                    const float* __restrict__ wc,       // packed [Wiou;Uiou]
                    const float* __restrict__ wf,       // packed [Wf;Uf]
                    float*       __restrict__ out,      // (S,B,H) ++ (B,H)
                    float*       __restrict__ hsum,     // (B,S+1,H) scratch
                    float*       __restrict__ fcsum)    // (B,S+1,H) scratch
{
    extern __shared__ float smem[];
    float* Xs  = smem;                     // [kBT][XS_LD]  A-operand, then g
    float* xfs = Xs  + kBT * XS_LD;        // [kBT][XFS_LD] gathered parent x
    float* hsb = xfs + kBT * XFS_LD;       // [kBT][HS_LD]  h
    float* csb = hsb + kBT * HS_LD;        // [kBT][HS_LD]  c
    float* fct = csb + kBT * HS_LD;        // [kBT][HS_LD]  f*c sum at node
    __shared__ int pbuf[kBT];              // parents of this step's 32 rows

    const int tid  = threadIdx.x;
    const int wave = tid >> 5;             // 0..15
    const int lane = tid & 31;
    const int col  = lane & 15;            // N / M-row index within tile
    const int half = lane >> 4;            // 0: lanes 0-15, 1: lanes 16-31
    const int b0   = blockIdx.x * kBT;     // this WG owns batch rows [b0, b0+32)

    for (int s = 0; s < kS; ++s) {
        if (tid < kBT) pbuf[tid] = parents[(b0 + tid) * kS + s];
        __syncthreads();

        // ---- Stage A = [inputs[s] | h_t] and gathered parent inputs -------
        for (int idx = tid; idx < kBT * (kD / 4); idx += kThreads) {
            int r = idx >> 7;              // kD/4 == 128
            int d = (idx & 127) * 4;
            *(float4*)&Xs[r * XS_LD + d] =
                *(const float4*)&inputs[((size_t)s * kB + (b0 + r)) * kD + d];
            int p = pbuf[r];
            float4 xv = make_float4(0.f, 0.f, 0.f, 0.f);
            if (p < kS)
                xv = *(const float4*)&inputs[((size_t)p * kB + (b0 + r)) * kD + d];
            *(float4*)&xfs[r * XFS_LD + d] = xv;
        }
        for (int idx = tid; idx < kBT * (kH / 4); idx += kThreads) {
            int r = idx >> 6;              // kH/4 == 64
            int h = (idx & 63) * 4;
            size_t base = ((size_t)(b0 + r) * (kS + 1) + s) * kH + h;
            *(float4*)&Xs[r * XS_LD + kD + h] = *(const float4*)&hsum[base];
            *(float4*)&fct[r * HS_LD + h]     = *(const float4*)&fcsum[base];
        }
        __syncthreads();

        // ---- g(32x768) = [x_s | h_t] @ [W_iou ; U_iou] (packed) -----------
        // wave w owns N-tiles n0, n0+256, n0+512 for BOTH M-tiles (rows 0-15
        // and 16-31); each B fragment load feeds two WMMAs.
        v8f acc00 = {}; v8f acc01 = {}; v8f acc02 = {};   // M-tile 0
        v8f acc10 = {}; v8f acc11 = {}; v8f acc12 = {};   // M-tile 1
        {
            const int n0 = wave * 16;
            const float* Xrow0 = &Xs[col * XS_LD];
            const float* Xrow1 = &Xs[(col + 16) * XS_LD];
            #pragma unroll 2
            for (int kb4 = 0; kb4 < kKC / 4; ++kb4) {
                v2f a0 = *(const v2f*)(Xrow0 + kb4 * 4 + 2 * half);
                v2f a1 = *(const v2f*)(Xrow1 + kb4 * 4 + 2 * half);
                const float* bp = wc + ((size_t)kb4 * kG + n0 + col) * 4 + 2 * half;
                v2f bq0 = *(const v2f*)(bp);
                v2f bq1 = *(const v2f*)(bp + 256 * 4);
                v2f bq2 = *(const v2f*)(bp + 512 * 4);
                acc00 = wmma_f32(a0, bq0, acc00);
                acc10 = wmma_f32(a1, bq0, acc10);
                acc01 = wmma_f32(a0, bq1, acc01);
                acc11 = wmma_f32(a1, bq1, acc11);
                acc02 = wmma_f32(a0, bq2, acc02);
                acc12 = wmma_f32(a1, bq2, acc12);
            }
        }
        __syncthreads();   // everyone finished reading Xs
        {
            // D layout: VGPR v, lanes 0-15 -> (M=v, N=lane); 16-31 -> (M=v+8)
            const int n0 = wave * 16;
            #pragma unroll
            for (int vr = 0; vr < 8; ++vr) {
                int m0 = vr + 8 * half;
                int m1 = m0 + 16;
                Xs[m0 * XS_LD + n0 + col]       = acc00[vr];
                Xs[m0 * XS_LD + 256 + n0 + col] = acc01[vr];
                Xs[m0 * XS_LD + 512 + n0 + col] = acc02[vr];
                Xs[m1 * XS_LD + n0 + col]       = acc10[vr];
                Xs[m1 * XS_LD + 256 + n0 + col] = acc11[vr];
                Xs[m1 * XS_LD + 512 + n0 + col] = acc12[vr];
            }
        }
        __syncthreads();

        // ---- gates, c, h (float4 over hidden dim) -------------------------
        for (int idx = tid; idx < kBT * (kH / 4); idx += kThreads) {
            int r = idx >> 6;
            int n = (idx & 63) * 4;
            float4 gi = *(const float4*)&Xs[r * XS_LD + n];
            float4 go = *(const float4*)&Xs[r * XS_LD + 256 + n];
            float4 gu = *(const float4*)&Xs[r * XS_LD + 512 + n];
            float4 bi = *(const float4*)&b_iou[n];
            float4 bo = *(const float4*)&b_iou[256 + n];
            float4 bu = *(const float4*)&b_iou[512 + n];
            float4 fcv = *(const float4*)&fct[r * HS_LD + n];
            float hv[4], cv[4];
            const float giv[4] = {gi.x + bi.x, gi.y + bi.y, gi.z + bi.z, gi.w + bi.w};
            const float gov[4] = {go.x + bo.x, go.y + bo.y, go.z + bo.z, go.w + bo.w};
            const float guv[4] = {gu.x + bu.x, gu.y + bu.y, gu.z + bu.z, gu.w + bu.w};
            const float fv[4]  = {fcv.x, fcv.y, fcv.z, fcv.w};
            #pragma unroll
            for (int j = 0; j < 4; ++j) {
                float c = sigmoidf(giv[j]) * tanhf(guv[j]) + fv[j];
                float h = sigmoidf(gov[j]) * tanhf(c);
                hv[j] = h; cv[j] = c;
            }
            float4 h4 = make_float4(hv[0], hv[1], hv[2], hv[3]);
            float4 c4 = make_float4(cv[0], cv[1], cv[2], cv[3]);
            *(float4*)&hsb[r * HS_LD + n] = h4;
            *(float4*)&csb[r * HS_LD + n] = c4;
            *(float4*)&out[((size_t)s * kB + (b0 + r)) * kH + n] = h4;  // contexts
            if (s == kS - 1)                                            // root_h
                *(float4*)&out[(size_t)kS * kB * kH + (size_t)(b0 + r) * kH + n] = h4;
        }
        __syncthreads();

        // ---- fpre(32x256) = x_parent @ W_f + h @ U_f ----------------------
        // wave w owns N-tile n0 for both M-tiles; B load shared by 2 WMMAs.
        v8f facc0 = {}; v8f facc1 = {};
        {
            const int n0 = wave * 16;
            const float* Arow0 = &xfs[col * XFS_LD];
            const float* Arow1 = &xfs[(col + 16) * XFS_LD];
            #pragma unroll 2
            for (int kb4 = 0; kb4 < kD / 4; ++kb4) {
                v2f a0 = *(const v2f*)(Arow0 + kb4 * 4 + 2 * half);
                v2f a1 = *(const v2f*)(Arow1 + kb4 * 4 + 2 * half);
                v2f b = *(const v2f*)(wf + ((size_t)kb4 * kH + n0 + col) * 4 + 2 * half);
                facc0 = wmma_f32(a0, b, facc0);
                facc1 = wmma_f32(a1, b, facc1);
            }
            const float* Hrow0 = &hsb[col * HS_LD];
            const float* Hrow1 = &hsb[(col + 16) * HS_LD];
            #pragma unroll 2
            for (int kb4 = kD / 4; kb4 < kKC / 4; ++kb4) {
                int kk = (kb4 - kD / 4) * 4 + 2 * half;
                v2f a0 = *(const v2f*)(Hrow0 + kk);
                v2f a1 = *(const v2f*)(Hrow1 + kk);
                v2f b = *(const v2f*)(wf + ((size_t)kb4 * kH + n0 + col) * 4 + 2 * half);
                facc0 = wmma_f32(a0, b, facc0);
                facc1 = wmma_f32(a1, b, facc1);
            }
        }
        // ---- f gate + scatter-add into parent accumulators ----------------
        {
            const int n0 = wave * 16;
            #pragma unroll
            for (int vr = 0; vr < 8; ++vr) {
                #pragma unroll
                for (int mt = 0; mt < 2; ++mt) {
                    int m = vr + 8 * half + 16 * mt;
                    float pv = (mt == 0) ? facc0[vr] : facc1[vr];
                    int p = pbuf[m];
                    float pre = pv + ((p < kS) ? b_f[n0 + col] : 0.0f);
                    float f  = sigmoidf(pre);
                    float hv = hsb[m * HS_LD + n0 + col];
                    float fc = f * csb[m * HS_LD + n0 + col];
                    size_t base = ((size_t)(b0 + m) * (kS + 1) + p) * kH + n0 + col;
                    unsafeAtomicAdd(&hsum[base], hv);
                    unsafeAtomicAdd(&fcsum[base], fc);
                }
            }
        }
        __threadfence();   // make scatter visible to our own next-step loads
        __syncthreads();
    }
}

extern "C" void kernel_launch(void* const* d_in, const int* in_sizes, int n_in,
                              void* d_out, int out_size, void* d_ws, size_t ws_size,
                              hipStream_t stream) {
    const float* inputs  = (const float*)d_in[0];
    const int*   parents = (const int*)  d_in[1];
    const float* W_iou   = (const float*)d_in[2];
    const float* b_iou   = (const float*)d_in[3];
    const float* U_iou   = (const float*)d_in[4];
    const float* W_f     = (const float*)d_in[5];
    const float* b_f     = (const float*)d_in[6];
    const float* U_f     = (const float*)d_in[7];
    float* out = (float*)d_out;

    const size_t nsum = (size_t)kB * (kS + 1) * kH;   // per accumulator
    float* hsum  = (float*)d_ws;
    float* fcsum = hsum + nsum;
    float* wc    = fcsum + nsum;                      // 192*768*4 floats
    float* wf    = wc + (size_t)(kKC / 4) * kG * 4;   // 192*256*4 floats

    zero_ws_kernel<<<2048, 256, 0, stream>>>((float4*)hsum, (2 * nsum) / 4);

    const int pack_items = (kKC / 4) * kG;            // 147456 (covers wf too)
    pack_weights_kernel<<<(pack_items + 255) / 256, 256, 0, stream>>>(
        W_iou, U_iou, W_f, U_f, wc, wf);

    const size_t lds_bytes =
        (size_t)(kBT * XS_LD + kBT * XFS_LD + 3 * kBT * HS_LD) * sizeof(float);
    tree_lstm_scan<<<kB / kBT, kThreads, lds_bytes, stream>>>(
        inputs, parents, b_iou, b_f, wc, wf, out, hsum, fcsum);
}